// LikelihoodOracleControls_74680891343521
// MI455X (gfx1250) — compile-verified
//
#include <hip/hip_runtime.h>
#include <hip/hip_bf16.h>
#include <math.h>

typedef float v2f __attribute__((ext_vector_type(2)));
typedef float v4f __attribute__((ext_vector_type(4)));
typedef float v8f __attribute__((ext_vector_type(8)));

#define TT 64
#define NB 8
#define KB 256
#define TS 512
#define CC 512

#define NEG_INF (-__builtin_inff())

// -----------------------------------------------------------------------------
// Kernel 1: z[m] = dot(x[m, :], w) + b for m in [0, TT*NB*KB), C = 512,
// using V_WMMA_F32_16X16X4_F32 (full fp32 matrix unit, no precision loss).
//
// One wave32 per 16-row tile. A-layout (ISA 7.12.2, 32-bit A 16x4):
//   lane 0-15 : M = lane,     VGPR0 = K0, VGPR1 = K1
//   lane 16-31: M = lane-16,  VGPR0 = K2, VGPR1 = K3
// We choose the k->c mapping per iteration pair so each lane issues ONE
// contiguous 16B load of x and one of w:
//   wmma #0: k0->c0, k1->c0+1, k2->c0+4, k3->c0+5
//   wmma #1: k0->c0+2,k1->c0+3,k2->c0+6, k3->c0+7
// B is w broadcast over all 16 N columns, so D[m, n] == z_partial[m] for all n.
// -----------------------------------------------------------------------------
__global__ void __launch_bounds__(256)
gemv_wmma_kernel(const float* __restrict__ x, const float* __restrict__ w,
                 const float* __restrict__ bias, float* __restrict__ z) {
  const int lane = threadIdx.x & 31;
  const int wave = (blockIdx.x * blockDim.x + threadIdx.x) >> 5;  // 8192 waves
  const int m0   = wave * 16;
  const int mrow = m0 + (lane & 15);
  const int hi   = lane >> 4;  // 0 or 1 (K half)

  const float* __restrict__ xp = x + (size_t)mrow * CC + 4 * hi;
  const float* __restrict__ wp = w + 4 * hi;

  v8f acc0 = {};
  v8f acc1 = {};
#pragma unroll 4
  for (int c0 = 0; c0 < CC; c0 += 8) {
    v4f xa = *(const v4f*)(xp + c0);  // x[m, c0+4*hi .. +3]
    v4f wb = *(const v4f*)(wp + c0);  // w[c0+4*hi .. +3]
    v2f a0; a0.x = xa.x; a0.y = xa.y;
    v2f a1; a1.x = xa.z; a1.y = xa.w;
    v2f b0; b0.x = wb.x; b0.y = wb.y;
    v2f b1; b1.x = wb.z; b1.y = wb.w;
    acc0 = __builtin_amdgcn_wmma_f32_16x16x4_f32(
        false, a0, false, b0, (short)0, acc0, false, false);
    acc1 = __builtin_amdgcn_wmma_f32_16x16x4_f32(
        false, a1, false, b1, (short)0, acc1, false, false);
  }
  const float bb = bias[0];
  // C/D layout: lane n (0-15) holds N=n for M=vgpr (lanes 0-15) / M=8+vgpr
  // (lanes 16-31). All N columns are identical; lanes 0 and 16 write rows.
  if ((lane & 15) == 0) {
    float* zp = z + m0 + 8 * hi;
    v4f o0, o1;
    o0.x = acc0[0] + acc1[0] + bb;
    o0.y = acc0[1] + acc1[1] + bb;
    o0.z = acc0[2] + acc1[2] + bb;
    o0.w = acc0[3] + acc1[3] + bb;
    o1.x = acc0[4] + acc1[4] + bb;
    o1.y = acc0[5] + acc1[5] + bb;
    o1.z = acc0[6] + acc1[6] + bb;
    o1.w = acc0[7] + acc1[7] + bb;
    *(v4f*)(zp)     = o0;
    *(v4f*)(zp + 4) = o1;
  }
}

__device__ __forceinline__ float log_sigmoid_f(float v) {
  // stable: v>=0 -> -log1p(exp(-v));  v<0 -> v - log1p(exp(v))
  float l = log1pf(expf(-fabsf(v)));
  return (v >= 0.0f) ? -l : (v - l);
}

// -----------------------------------------------------------------------------
// Kernel 2: one thread per (t, s) cell (64*512). Binary-search s in the sorted
// indices[t, 0:256]; fill E (with -inf default) and Cg (read=0 /
// write = (s>t ? 0 : -inf) default; scattered controls where found).
// -----------------------------------------------------------------------------
__global__ void __launch_bounds__(256)
scatter_kernel(const float* __restrict__ emissions, const int* __restrict__ indices,
               const float* __restrict__ z, float* __restrict__ Eo,
               float* __restrict__ Cgo) {
  int cell = blockIdx.x * blockDim.x + threadIdx.x;
  if (cell >= TT * TS) return;
  const int t = cell >> 9;   // / TS
  const int s = cell & (TS - 1);

  const int* __restrict__ row = indices + t * KB;
  int lo = 0, hiB = KB;
  while (lo < hiB) {
    int mid = (lo + hiB) >> 1;
    if (row[mid] < s) lo = mid + 1; else hiB = mid;
  }
  const bool found = (lo < KB) && (row[lo] == s);
  const int k = lo;
  const float tri_w = (s > t) ? 0.0f : NEG_INF;

#pragma unroll
  for (int n = 0; n < NB; ++n) {
    const int r = t * NB + n;                 // row id in (Tt*N)
    float e = NEG_INF, rd = 0.0f, wr = tri_w;
    if (found) {
      e = emissions[r * KB + k];
      float zv = z[r * KB + k];
      float sv = log_sigmoid_f(zv);
      rd = sv;
      wr = sv - zv;
    }
    Eo[r * TS + s] = e;
    if (t < TT - 1) {
      float* cp = Cgo + ((size_t)(r * TS + s)) * 2;
      cp[0] = rd;
      cp[1] = wr;
    }
  }
}

// -----------------------------------------------------------------------------
// Kernel 3: one 64-thread block per (t, n) row. winmax over next 8 entries,
// gamma_raw = !(E[s] < winmax), normalize over the row, emit gamma and
// (for t>=1) write = gamma, read = 1 - gamma.
// -----------------------------------------------------------------------------
__global__ void __launch_bounds__(64)
gamma_kernel(const float* __restrict__ Eo, float* __restrict__ gam,
             float* __restrict__ rd, float* __restrict__ wr) {
  __shared__ float rowbuf[TS];
  __shared__ int cnt;
  const int r = blockIdx.x;      // 0..511 == t*NB+n
  const int tid = threadIdx.x;   // 0..63
  if (tid == 0) cnt = 0;

  const float* __restrict__ erow = Eo + r * TS;
#pragma unroll
  for (int i = tid; i < TS; i += 64) rowbuf[i] = erow[i];
  __syncthreads();

  float g[8];
  int local = 0;
  const int s0 = tid * 8;
#pragma unroll
  for (int j = 0; j < 8; ++j) {
    const int s = s0 + j;
    float wm = NEG_INF;
#pragma unroll
    for (int d = 1; d <= 8; ++d) {
      const int ss = s + d;
      const float v = (ss < TS) ? rowbuf[ss] : NEG_INF;
      wm = fmaxf(wm, v);
    }
    // (-inf < -inf) is false, matching jnp semantics exactly
    const float gi = (rowbuf[s] < wm) ? 0.0f : 1.0f;
    g[j] = gi;
    local += (int)gi;
  }
  atomicAdd(&cnt, local);
  __syncthreads();

  const float inv = 1.0f / (float)cnt;
  const int t = r >> 3;  // / NB
#pragma unroll
  for (int j = 0; j < 8; ++j) {
    const int s = s0 + j;
    const float gv = g[j] * inv;
    gam[r * TS + s] = gv;
    if (t >= 1) {
      const int ro = r - NB;  // (t-1)*NB + n
      wr[ro * TS + s] = gv;
      rd[ro * TS + s] = 1.0f - gv;
    }
  }
}

extern "C" void kernel_launch(void* const* d_in, const int* in_sizes, int n_in,
                              void* d_out, int out_size, void* d_ws, size_t ws_size,
                              hipStream_t stream) {
  const float* x         = (const float*)d_in[0];  // (64, 8, 256, 512)
  const float* emissions = (const float*)d_in[1];  // (64, 8, 256)
  const int*   indices   = (const int*)d_in[2];    // (64, 256) sorted
  // d_in[3] = src_length (512), d_in[4] = src_mask (unused by reference)
  const float* gate_w    = (const float*)d_in[5];  // (1, 512)
  const float* gate_b    = (const float*)d_in[6];  // (1,)

  float* out = (float*)d_out;
  // Output layout (flat, return order):
  float* E_out     = out;                                   // 64*8*512
  float* gamma_out = E_out + TT * NB * TS;                  // 64*8*512
  float* Cg_out    = gamma_out + TT * NB * TS;              // 63*8*512*2
  float* read_out  = Cg_out + (TT - 1) * NB * TS * 2;       // 63*8*512
  float* write_out = read_out + (TT - 1) * NB * TS;         // 63*8*512

  float* zbuf = (float*)d_ws;  // TT*NB*KB floats = 512 KB scratch

  // 1) z = x . w + b  (8192 wave-tiles of 16 rows; 256 thr = 8 waves/block)
  gemv_wmma_kernel<<<dim3((TT * NB * KB / 16) / 8), dim3(256), 0, stream>>>(
      x, gate_w, gate_b, zbuf);

  // 2) scatter into E grid and Cg grid (one thread per (t, s))
  scatter_kernel<<<dim3((TT * TS) / 256), dim3(256), 0, stream>>>(
      emissions, indices, zbuf, E_out, Cg_out);

  // 3) windowed max -> gamma, read, write (one block per (t, n) row)
  gamma_kernel<<<dim3(TT * NB), dim3(64), 0, stream>>>(
      E_out, gamma_out, read_out, write_out);
}